// SmoothingLayer_22651657519653
// MI455X (gfx1250) — compile-verified
//
#include <hip/hip_runtime.h>
#include <cmath>

// ---------------------------------------------------------------------------
// 3D separable Gaussian smoothing (sigma=2, truncate=3 -> 13 taps) on
// [B=2, X=128, Y=128, Z=128, C=8] fp32, NDHWC layout (C innermost).
// Each 1-D pass is a banded matrix multiply on the matrix pipe:
//   Out[16x16] = Band[16x28] x InExt[28x16]   via 7x V_WMMA_F32_16X16X4_F32.
// Halo handling is branch-free: clamped-address load + cndmask select.
// ---------------------------------------------------------------------------

typedef float v2f __attribute__((ext_vector_type(2)));
typedef float v8f __attribute__((ext_vector_type(8)));

struct GaussW { float w[13]; };

__device__ __forceinline__ int iclamp(int v, int lo, int hi) {
    return v < lo ? lo : (v > hi ? hi : v);
}

// Strides (elements): C:1, Z:8, Y:1024, X:131072, B:16777216. Axis length 128.
// PASS 0: conv along X (stride 131072), lines = (B, Y*Z*C)
// PASS 1: conv along Y (stride 1024),   lines = (B, X, Z*C)
// PASS 2: conv along Z (stride 8),      lines = (B, X, Y, C)
template <int PASS>
__global__ __launch_bounds__(256) void gauss_wmma_pass(const float* __restrict__ in,
                                                       float* __restrict__ out,
                                                       GaussW gw) {
    // Zero-padded band-weight table: wpad[16+d] = w[d] for d in [0,12], else 0.
    __shared__ float wpad[64];
    const unsigned tid = threadIdx.x;
    if (tid < 64) wpad[tid] = 0.0f;
    __syncthreads();
    if (tid < 13) wpad[16 + tid] = gw.w[tid];
    __syncthreads();

    const unsigned wave   = tid >> 5;
    const unsigned lane   = tid & 31;
    const unsigned laneLo = lane & 15;   // M for A-frags, N for B/C/D-frags
    const unsigned laneHi = lane >> 4;   // half-wave selector

    // One wave32 per 16x16 tile. tiles = lineGroups(16384) * axisBlocks(8).
    const unsigned tile = blockIdx.x * 8u + wave;
    const unsigned ab   = tile & 7u;          // axis block: outputs [16*ab, 16*ab+15]
    const unsigned lg   = tile >> 3;          // line group (16 lines)
    const unsigned n    = lg * 16u + laneLo;  // this lane's line index

    // Base memory offset of line n (elements) and the along-axis stride.
    unsigned base;
    unsigned S;
    if (PASS == 0) {
        S = 131072u;
        base = (n & 131071u) + (n >> 17) * 16777216u;
    } else if (PASS == 1) {
        S = 1024u;
        base = (n & 1023u) + ((n >> 10) & 127u) * 131072u + (n >> 17) * 16777216u;
    } else {
        S = 8u;
        base = (n & 7u) + ((n >> 3) & 127u) * 1024u +
               ((n >> 10) & 127u) * 131072u + (n >> 17) * 16777216u;
    }

    const int m0 = (int)(ab * 16u);  // first output position along the axis

    v8f acc = {};  // conv has no bias -> accumulate from zero
#pragma unroll
    for (int c = 0; c < 7; ++c) {
        const int k0 = 4 * c + 2 * (int)laneHi;  // K indices k0, k0+1 for this lane

        // A fragment: Band[m=laneLo][k] = wpad[16 + k - m]  (zero outside band)
        v2f a;
        a.x = wpad[16 + k0 - (int)laneLo];
        a.y = wpad[16 + k0 + 1 - (int)laneLo];

        // B fragment: InExt[k][n] = in[base + (m0 + k - 6)*S], zero outside [0,128).
        // Branch-free: load from clamped (valid) address, select 0 if out of range.
        const int p0 = m0 + k0 - 6;
        const int p1 = p0 + 1;
        const unsigned q0 = (unsigned)iclamp(p0, 0, 127);
        const unsigned q1 = (unsigned)iclamp(p1, 0, 127);
        const float v0 = in[base + q0 * S];
        const float v1 = in[base + q1 * S];
        v2f b;
        b.x = ((unsigned)p0 < 128u) ? v0 : 0.0f;
        b.y = ((unsigned)p1 < 128u) ? v1 : 0.0f;

        // D = A x B + C   (fp32, 16x16x4)
        acc = __builtin_amdgcn_wmma_f32_16x16x4_f32(
            /*neg_a=*/false, a, /*neg_b=*/false, b,
            /*c_mod=*/(short)0, acc, /*reuse_a=*/false, /*reuse_b=*/false);
    }

    // D layout: acc[v] holds Out[m0 + v + 8*laneHi][n]
#pragma unroll
    for (int v = 0; v < 8; ++v) {
        const int m = m0 + v + 8 * (int)laneHi;
        out[base + (unsigned)m * S] = acc[v];
    }
}

static void make_weights(GaussW& g) {
    // Match the numpy fp32 computation: exp(-0.5*x^2/sigma^2), normalized.
    float k[13];
    float s = 0.0f;
    for (int i = 0; i < 13; ++i) {
        const float x = (float)(i - 6);
        k[i] = expf(-0.5f * x * x / 4.0f);
    }
    for (int i = 0; i < 13; ++i) s += k[i];
    for (int i = 0; i < 13; ++i) g.w[i] = k[i] / s;
}

extern "C" void kernel_launch(void* const* d_in, const int* in_sizes, int n_in,
                              void* d_out, int out_size, void* d_ws, size_t ws_size,
                              hipStream_t stream) {
    (void)in_sizes; (void)n_in; (void)out_size; (void)ws_size;

    const float* in  = (const float*)d_in[0];
    float*       out = (float*)d_out;
    float*       tmp = (float*)d_ws;  // one 128 MiB ping-pong buffer

    GaussW gw;
    make_weights(gw);

    // Per pass: 262144 lines / 16 = 16384 line groups x 8 axis blocks
    //         = 131072 tiles; 8 waves (tiles) per 256-thread block.
    const int blocks = 16384;
    const dim3 blk(256);

    gauss_wmma_pass<0><<<blocks, blk, 0, stream>>>(in,  out, gw);  // X pass
    gauss_wmma_pass<1><<<blocks, blk, 0, stream>>>(out, tmp, gw);  // Y pass
    gauss_wmma_pass<2><<<blocks, blk, 0, stream>>>(tmp, out, gw);  // Z pass
}